// CrystalHeads_85667417686346
// MI455X (gfx1250) — compile-verified
//
#include <hip/hip_runtime.h>
#include <hip/hip_bf16.h>
#include <math.h>

#define D_MODEL 512
#define BATCH   4
#define NATOM   2048
#define VZ      101
#define M_TOT   (BATCH * NATOM)   // 8192

typedef __attribute__((ext_vector_type(2))) float v2f;
typedef __attribute__((ext_vector_type(8))) float v8f;

static __device__ __forceinline__ v8f wmma4(v2f a, v2f b, v8f c) {
  // D = A(16x4 f32) * B(4x16 f32) + C(16x16 f32)
  return __builtin_amdgcn_wmma_f32_16x16x4_f32(
      /*neg_a=*/false, a, /*neg_b=*/false, b,
      /*c_mod=*/(short)0, c, /*reuse_a=*/false, /*reuse_b=*/false);
}

// ---------------------------------------------------------------------------
// Kernel 1: fused q/k projection.  out layout: qk[row][0..511]=q, [512..1023]=k
// GEMM: M=8192 (h_atoms rows), N=1024, K=512.  One wave -> 32x64 tile
// (two 16-row A fragments share each B fragment).
// ---------------------------------------------------------------------------
__global__ __launch_bounds__(32) void k_proj_qk(
    const float* __restrict__ h,
    const float* __restrict__ Wq, const float* __restrict__ bq,
    const float* __restrict__ Wk, const float* __restrict__ bk,
    float* __restrict__ qk)
{
  const int lane = threadIdx.x;
  const int half = lane >> 4;
  const int ln   = lane & 15;
  const int i0   = blockIdx.y * 32;   // global row tile base (0..8191)
  const int c0   = blockIdx.x * 64;   // combined col group (0..1023)

  // A fragment sources: rows i0+ln and i0+16+ln of h_atoms (skip h_lat rows).
  const float* aPtr[2];
#pragma unroll
  for (int g = 0; g < 2; ++g) {
    const int r  = i0 + 16 * g + ln;
    const int b  = r >> 11;
    const int ia = r & (NATOM - 1);
    aPtr[g] = h + ((size_t)b * (NATOM + 1) + ia) * D_MODEL + 2 * half;
  }

  const float* W    = (c0 < D_MODEL) ? Wq : Wk;
  const float* bias = (c0 < D_MODEL) ? bq : bk;
  const int cc0 = c0 & (D_MODEL - 1);

  const float* bPtr[4];
#pragma unroll
  for (int t = 0; t < 4; ++t)
    bPtr[t] = W + (size_t)(2 * half) * D_MODEL + (cc0 + 16 * t + ln);

  v8f acc[2][4] = {};
  for (int k0 = 0; k0 < D_MODEL; k0 += 4) {
    v2f a0 = *(const v2f*)(aPtr[0] + k0);       // h[m][k0+2h..k0+2h+1]
    v2f a1 = *(const v2f*)(aPtr[1] + k0);
#pragma unroll
    for (int t = 0; t < 4; ++t) {
      v2f bf;
      bf.x = bPtr[t][(size_t)k0 * D_MODEL];        // W[k0+2h][c]
      bf.y = bPtr[t][(size_t)(k0 + 1) * D_MODEL];  // W[k0+2h+1][c]
      acc[0][t] = wmma4(a0, bf, acc[0][t]);
      acc[1][t] = wmma4(a1, bf, acc[1][t]);
    }
  }

#pragma unroll
  for (int t = 0; t < 4; ++t) {
    const int c  = c0 + 16 * t + ln;
    const float bv = bias[cc0 + 16 * t + ln];
#pragma unroll
    for (int g = 0; g < 2; ++g) {
#pragma unroll
      for (int rr = 0; rr < 8; ++rr) {
        const int m = 16 * g + rr + (half ? 8 : 0);
        qk[(size_t)(i0 + m) * 1024 + c] = acc[g][t][rr] + bv;
      }
    }
  }
}

// ---------------------------------------------------------------------------
// Kernel 2: scores = (q @ k^T) * D^-0.5 per batch.  One wave -> 32x64 tile.
// ---------------------------------------------------------------------------
__global__ __launch_bounds__(32) void k_scores(
    const float* __restrict__ qk, float* __restrict__ scores)
{
  const int lane = threadIdx.x;
  const int half = lane >> 4;
  const int ln   = lane & 15;
  const int b  = blockIdx.z;
  const int i0 = blockIdx.y * 32;
  const int j0 = blockIdx.x * 64;

  const float* qb = qk + (size_t)b * NATOM * 1024;
  const float* aPtr[2];
#pragma unroll
  for (int g = 0; g < 2; ++g)
    aPtr[g] = qb + (size_t)(i0 + 16 * g + ln) * 1024 + 2 * half;   // q rows

  const float* bPtr[4];                                            // k rows (B = k^T)
#pragma unroll
  for (int t = 0; t < 4; ++t)
    bPtr[t] = qb + (size_t)(j0 + 16 * t + ln) * 1024 + D_MODEL + 2 * half;

  v8f acc[2][4] = {};
  for (int k0 = 0; k0 < D_MODEL; k0 += 4) {
    v2f a0 = *(const v2f*)(aPtr[0] + k0);
    v2f a1 = *(const v2f*)(aPtr[1] + k0);
#pragma unroll
    for (int t = 0; t < 4; ++t) {
      v2f bf = *(const v2f*)(bPtr[t] + k0);
      acc[0][t] = wmma4(a0, bf, acc[0][t]);
      acc[1][t] = wmma4(a1, bf, acc[1][t]);
    }
  }

  const float scale = 0.04419417382415922f;  // 512^-0.5
  float* srow = scores + ((size_t)b * NATOM + i0) * NATOM + j0;
#pragma unroll
  for (int t = 0; t < 4; ++t) {
#pragma unroll
    for (int g = 0; g < 2; ++g) {
#pragma unroll
      for (int rr = 0; rr < 8; ++rr) {
        const int m = 16 * g + rr + (half ? 8 : 0);
        srow[(size_t)m * NATOM + 16 * t + ln] = acc[g][t][rr] * scale;
      }
    }
  }
}

// ---------------------------------------------------------------------------
// Kernel 3: type_logits = h_atoms @ Wt + bt  (N=101 -> 7 col tiles, guarded)
// One wave -> 32x112 tile.
// ---------------------------------------------------------------------------
__global__ __launch_bounds__(32) void k_type(
    const float* __restrict__ h, const float* __restrict__ Wt,
    const float* __restrict__ bt, float* __restrict__ out)
{
  const int lane = threadIdx.x;
  const int half = lane >> 4;
  const int ln   = lane & 15;
  const int i0 = blockIdx.x * 32;

  const float* aPtr[2];
#pragma unroll
  for (int g = 0; g < 2; ++g) {
    const int r  = i0 + 16 * g + ln;
    const int b  = r >> 11;
    const int ia = r & (NATOM - 1);
    aPtr[g] = h + ((size_t)b * (NATOM + 1) + ia) * D_MODEL + 2 * half;
  }

  const float* bPtr[7];
#pragma unroll
  for (int t = 0; t < 7; ++t) {
    int c = 16 * t + ln;
    if (c > VZ - 1) c = VZ - 1;  // clamp loads in-bounds; stores are guarded
    bPtr[t] = Wt + (size_t)(2 * half) * VZ + c;
  }

  v8f acc[2][7] = {};
  for (int k0 = 0; k0 < D_MODEL; k0 += 4) {
    v2f a0 = *(const v2f*)(aPtr[0] + k0);
    v2f a1 = *(const v2f*)(aPtr[1] + k0);
#pragma unroll
    for (int t = 0; t < 7; ++t) {
      v2f bf;
      bf.x = bPtr[t][(size_t)k0 * VZ];
      bf.y = bPtr[t][(size_t)(k0 + 1) * VZ];
      acc[0][t] = wmma4(a0, bf, acc[0][t]);
      acc[1][t] = wmma4(a1, bf, acc[1][t]);
    }
  }

#pragma unroll
  for (int t = 0; t < 7; ++t) {
    const int c = 16 * t + ln;
    if (c < VZ) {
      const float bv = bt[c];
#pragma unroll
      for (int g = 0; g < 2; ++g) {
#pragma unroll
        for (int rr = 0; rr < 8; ++rr) {
          const int m = 16 * g + rr + (half ? 8 : 0);
          out[(size_t)(i0 + m) * VZ + c] = acc[g][t][rr] + bv;
        }
      }
    }
  }
}

// ---------------------------------------------------------------------------
// Reductions
// ---------------------------------------------------------------------------
static __device__ __forceinline__ float waveSum(float v) {
#pragma unroll
  for (int o = 16; o > 0; o >>= 1) v += __shfl_xor(v, o, 32);
  return v;
}
static __device__ __forceinline__ float waveMax(float v) {
#pragma unroll
  for (int o = 16; o > 0; o >>= 1) v = fmaxf(v, __shfl_xor(v, o, 32));
  return v;
}
static __device__ __forceinline__ float blockSum(float v, float* sm, int wid, int lid) {
  v = waveSum(v);
  __syncthreads();
  if (lid == 0) sm[wid] = v;
  __syncthreads();
  float t = 0.f;
#pragma unroll
  for (int w = 0; w < 8; ++w) t += sm[w];
  return t;
}
static __device__ __forceinline__ float blockMax(float v, float* sm, int wid, int lid) {
  v = waveMax(v);
  __syncthreads();
  if (lid == 0) sm[wid] = v;
  __syncthreads();
  float t = sm[0];
#pragma unroll
  for (int w = 1; w < 8; ++w) t = fmaxf(t, sm[w]);
  return t;
}

// ---------------------------------------------------------------------------
// Kernel 4: row softmax + periodic-delta aggregation -> coord_vel
// One 256-thread block per (b,i) row; 2048 scores held in registers.
// ---------------------------------------------------------------------------
__global__ __launch_bounds__(256) void k_softmax_coord(
    const float* __restrict__ scores, const float* __restrict__ frac,
    const unsigned char* __restrict__ pad, float* __restrict__ coord_out)
{
  __shared__ float sm[8];
  const int row = blockIdx.x;           // 0..8191
  const int b   = row >> 11;
  const int i   = row & (NATOM - 1);
  const int tid = threadIdx.x;
  const int wid = tid >> 5, lid = tid & 31;

  const float* srow = scores + (size_t)row * NATOM;
  const unsigned char* pm = pad + (size_t)b * NATOM;

  float s[8];
  bool  mk[8];
#pragma unroll
  for (int u = 0; u < 8; ++u) {
    const int j = tid + 256 * u;
    s[u]  = srow[j];
    mk[u] = pm[j] != 0;
  }

  float mx = -3.402823466e38f;
#pragma unroll
  for (int u = 0; u < 8; ++u) if (!mk[u]) mx = fmaxf(mx, s[u]);
  mx = blockMax(mx, sm, wid, lid);

  float p[8], psum = 0.f;
#pragma unroll
  for (int u = 0; u < 8; ++u) {
    p[u] = mk[u] ? 0.f : __expf(s[u] - mx);
    psum += p[u];
  }
  const float tot = blockSum(psum, sm, wid, lid);

  const float* fb = frac + (size_t)b * NATOM * 3;
  const float fi0 = fb[(size_t)i * 3 + 0];
  const float fi1 = fb[(size_t)i * 3 + 1];
  const float fi2 = fb[(size_t)i * 3 + 2];

  float a0 = 0.f, a1 = 0.f, a2 = 0.f;
#pragma unroll
  for (int u = 0; u < 8; ++u) {
    const int j = tid + 256 * u;
    const float* fj = fb + (size_t)j * 3;
    float d0 = fi0 - fj[0]; d0 -= rintf(d0);   // rintf == round-half-even
    float d1 = fi1 - fj[1]; d1 -= rintf(d1);
    float d2 = fi2 - fj[2]; d2 -= rintf(d2);
    a0 += p[u] * d0;
    a1 += p[u] * d1;
    a2 += p[u] * d2;
  }
  const float A0 = blockSum(a0, sm, wid, lid);
  const float A1 = blockSum(a1, sm, wid, lid);
  const float A2 = blockSum(a2, sm, wid, lid);

  if (tid == 0) {
    const float den = fmaxf(tot, 1e-8f);
    const bool qm = pm[i] != 0;   // query-side pad mask
    float* co = coord_out + (size_t)row * 3;
    co[0] = qm ? 0.f : A0 / den;
    co[1] = qm ? 0.f : A1 / den;
    co[2] = qm ? 0.f : A2 / den;
  }
}

// ---------------------------------------------------------------------------
// Kernel 5: lattice_vel = h_lat @ Wl + bl   (4x6 outputs)
// ---------------------------------------------------------------------------
__global__ __launch_bounds__(256) void k_lattice(
    const float* __restrict__ h, const float* __restrict__ Wl,
    const float* __restrict__ bl, float* __restrict__ out)
{
  __shared__ float sm[8];
  const int b = blockIdx.x, tid = threadIdx.x;
  const int wid = tid >> 5, lid = tid & 31;
  const float* hl = h + ((size_t)b * (NATOM + 1) + NATOM) * D_MODEL;
  for (int c = 0; c < 6; ++c) {
    float p = 0.f;
    for (int d = tid; d < D_MODEL; d += 256) p += hl[d] * Wl[(size_t)d * 6 + c];
    const float t = blockSum(p, sm, wid, lid);
    if (tid == 0) out[b * 6 + c] = t + bl[c];
  }
}

// ---------------------------------------------------------------------------
extern "C" void kernel_launch(void* const* d_in, const int* in_sizes, int n_in,
                              void* d_out, int out_size, void* d_ws, size_t ws_size,
                              hipStream_t stream)
{
  const float* h  = (const float*)d_in[0];
  const float* fc = (const float*)d_in[1];
  const unsigned char* pad = (const unsigned char*)d_in[2];
  const float* Wt = (const float*)d_in[3];
  const float* bt = (const float*)d_in[4];
  const float* Wq = (const float*)d_in[5];
  const float* bq = (const float*)d_in[6];
  const float* Wk = (const float*)d_in[7];
  const float* bk = (const float*)d_in[8];
  const float* Wl = (const float*)d_in[9];
  const float* bl = (const float*)d_in[10];

  float* out_type  = (float*)d_out;                       // 4*2048*101
  float* out_coord = out_type + (size_t)M_TOT * VZ;       // 4*2048*3
  float* out_lat   = out_coord + (size_t)M_TOT * 3;       // 24

  float* qk     = (float*)d_ws;                           // 8192*1024 f32
  float* scores = qk + (size_t)M_TOT * 1024;              // 8192*2048 f32

  k_proj_qk<<<dim3(16, 256), 32, 0, stream>>>(h, Wq, bq, Wk, bk, qk);
  k_scores<<<dim3(32, 64, 4), 32, 0, stream>>>(qk, scores);
  k_type<<<dim3(256), 32, 0, stream>>>(h, Wt, bt, out_type);
  k_softmax_coord<<<dim3(8192), 256, 0, stream>>>(scores, fc, pad, out_coord);
  k_lattice<<<dim3(4), 256, 0, stream>>>(h, Wl, bl, out_lat);
}